// LearnablePixelwiseAnisoJBU_NoParent_22428319219928
// MI455X (gfx1250) — compile-verified
//
#include <hip/hip_runtime.h>
#include <hip/hip_bf16.h>
#include <cmath>

// ---------------------------------------------------------------------------
// LearnablePixelwiseAnisoJBU for MI455X (gfx1250)
//
// Roofline: output write 256MB dominates (-> ~11us @ 23.3TB/s). Grid (12.6MB)
// is L2-resident. Slicing is recast per 16x16 output tile as a dense f32 GEMM
//   D[128ch x 256px] = A[128 x 108cells] x W[108 x 256px]
// (108 = 3x3 low-res xy footprint x 12 z bins, W sparse 8-nnz per column),
// executed with V_WMMA_F32_16X16X4_F32. W fragments are register-cached per
// pixel-block (reused by all 8 channel blocks), A fragments are preloaded as
// one ds_load clause per job so the 27-WMMA chain runs without per-step LDS
// waits. A is staged into LDS with global_load_async_to_lds_b128 (ASYNCcnt).
// ---------------------------------------------------------------------------

typedef float v2f __attribute__((ext_vector_type(2)));
typedef float v8f __attribute__((ext_vector_type(8)));

#define B_     2
#define C_     128
#define HL     32
#define WL     32
#define HH     512
#define WH     512
#define NB     12          // num z bins
#define NCELL  (NB * 9)    // 108 cells per tile footprint
#define ASTR   136         // A row stride in floats (bank-conflict padding)
#define KSTEPS (NCELL / 4) // 27 wmma k-steps

// ---------------------------------------------------------------------------
// Kernel 1: antialiased bilinear (triangle, radius 16) downsample of guide
// luma 512x512 -> 32x32 (matches jax.image.resize bilinear w/ antialias).
// ---------------------------------------------------------------------------
__global__ void luma_lr_kernel(const float* __restrict__ guide,
                               float* __restrict__ luma_lr) {
  int idx = blockIdx.x * blockDim.x + threadIdx.x;
  if (idx >= B_ * HL * WL) return;
  int b = idx >> 10, rem = idx & 1023;
  int yl = rem >> 5, xl = rem & 31;
  float cy = (yl + 0.5f) * 16.0f - 0.5f;
  float cx = (xl + 0.5f) * 16.0f - 0.5f;
  int ys = (int)ceilf(cy - 16.0f); if (ys < 0) ys = 0;
  int ye = (int)floorf(cy + 16.0f); if (ye > HH - 1) ye = HH - 1;
  int xs = (int)ceilf(cx - 16.0f); if (xs < 0) xs = 0;
  int xe = (int)floorf(cx + 16.0f); if (xe > WH - 1) xe = WH - 1;
  const float* g = guide + (size_t)b * 3 * HH * WH;
  float acc = 0.0f, wsum = 0.0f;
  for (int y = ys; y <= ye; ++y) {
    float wy = 1.0f - fabsf((float)y - cy) * (1.0f / 16.0f);
    if (wy <= 0.0f) continue;
    for (int x = xs; x <= xe; ++x) {
      float wx = 1.0f - fabsf((float)x - cx) * (1.0f / 16.0f);
      if (wx <= 0.0f) continue;
      float l = (g[y * WH + x] + g[HH * WH + y * WH + x] +
                 g[2 * HH * WH + y * WH + x]) * (1.0f / 3.0f);
      acc += wy * wx * l;
      wsum += wy * wx;
    }
  }
  luma_lr[idx] = acc / wsum;
}

// ---------------------------------------------------------------------------
// Kernel 2: build bilateral grid, 3x3 zero-padded box blur (/9), scale by
// (1 + 0.01*(sx+sy)).  Output layout: gridF[b][z][y][x][c] (c fastest) so the
// slice kernel can stage whole cells contiguously; gridW[b][z][y][x].
// ---------------------------------------------------------------------------
__global__ void grid_kernel(const float* __restrict__ feat,
                            const float* __restrict__ luma_lr,
                            const float* __restrict__ sx_raw,
                            const float* __restrict__ sy_raw,
                            const float* __restrict__ sr_raw,
                            float* __restrict__ gridF,
                            float* __restrict__ gridW) {
  int bz = blockIdx.x;            // B*NB*HL blocks
  int y = bz & 31;
  int z = (bz >> 5) % NB;
  int b = bz / (NB * HL);
  int c = threadIdx.x;            // 128 threads = channel
  const float* lum = luma_lr + b * HL * WL;
  const float* fp = feat + (((size_t)b * C_ + c) * HL) * WL;
  for (int x = 0; x < WL; ++x) {
    float acc = 0.0f, wsum = 0.0f;
    for (int dy = -1; dy <= 1; ++dy) {
      int yn = y + dy;
      if (yn < 0 || yn > HL - 1) continue;
      for (int dx = -1; dx <= 1; ++dx) {
        int xn = x + dx;
        if (xn < 0 || xn > WL - 1) continue;
        float sr = expf(sr_raw[yn * WL + xn]);
        float wn = 1.0f - fabsf(lum[yn * WL + xn] * (float)(NB - 1) - (float)z)
                          / (sr * (float)NB + 1e-5f);
        wn = fmaxf(wn, 0.0f);
        acc += fp[yn * WL + xn] * wn;
        wsum += wn;
      }
    }
    float scale = (1.0f + 0.01f * (expf(sx_raw[y * WL + x]) +
                                   expf(sy_raw[y * WL + x]))) * (1.0f / 9.0f);
    size_t cell = (((size_t)b * NB + z) * HL + y) * WL + x;
    gridF[cell * C_ + c] = acc * scale;
    if (c == 0) gridW[cell] = wsum * scale;
  }
}

// ---------------------------------------------------------------------------
// Kernel 3: slice. One block = one 16x16 high-res tile of one batch.
// LDS: A[108][136] grid values (ch fastest), W[16 px-blocks][108][16] weights,
// invDen[256]. Each wave owns 2 pixel-blocks: register-cache that block's
// 27 W fragments, then per channel block preload 27 A fragments (one ds
// clause) and run the 27-WMMA chain back-to-back.
// ---------------------------------------------------------------------------
__global__ void __launch_bounds__(256)
slice_kernel(const float* __restrict__ guide,
             const float* __restrict__ gridF,
             const float* __restrict__ gridW,
             float* __restrict__ out) {
  extern __shared__ float smem[];
  float* A = smem;                       // NCELL * ASTR
  float* W = A + NCELL * ASTR;           // 16 * NCELL * 16
  float* invDen = W + 16 * NCELL * 16;   // 256

  const int tx = blockIdx.x, ty = blockIdx.y, b = blockIdx.z;
  const int tid = threadIdx.x;
  const int lane = tid & 31, wave = tid >> 5;
  const int x0h = tx * 16, y0h = ty * 16;

  const float s = (float)(WL - 1) / (float)(WH - 1);   // 31/511
  const int xlo = (int)floorf((float)x0h * s);
  const int ylo = (int)floorf((float)y0h * s);

  // ---- per-pixel coords & trilinear weights -------------------------------
  const int px = x0h + (tid & 15);
  const int py = y0h + (tid >> 4);
  const float* g = guide + (size_t)b * 3 * HH * WH;
  float luma = (g[py * WH + px] + g[HH * WH + py * WH + px] +
                g[2 * HH * WH + py * WH + px]) * (1.0f / 3.0f);
  float u = (float)px * s;
  float v = (float)py * s;
  float w = fminf(fmaxf(luma * (float)(NB - 1), 0.0f), (float)(NB - 1));
  int x0 = (int)u; float fx = u - (float)x0;
  int y0 = (int)v; float fy = v - (float)y0;
  int z0 = (int)w; if (z0 > NB - 1) z0 = NB - 1;
  float fz = w - (float)z0;
  int z1 = z0 + 1 < NB ? z0 + 1 : NB - 1;
  int lx = x0 - xlo, ly = y0 - ylo;       // in {0,1}
  float wx0 = 1.0f - fx, wx1 = fx;
  float wy0 = 1.0f - fy, wy1 = fy;
  float wz0 = 1.0f - fz, wz1 = fz;

  // ---- denominator (weight channel) via direct L2 gathers -----------------
  {
    int x1g = x0 + 1 < WL ? x0 + 1 : WL - 1;
    int y1g = y0 + 1 < HL ? y0 + 1 : HL - 1;
    const float* gw = gridW + (size_t)b * NB * HL * WL;
    float den =
        wz0 * (wy0 * (wx0 * gw[(z0 * HL + y0) * WL + x0] +
                      wx1 * gw[(z0 * HL + y0) * WL + x1g]) +
               wy1 * (wx0 * gw[(z0 * HL + y1g) * WL + x0] +
                      wx1 * gw[(z0 * HL + y1g) * WL + x1g])) +
        wz1 * (wy0 * (wx0 * gw[(z1 * HL + y0) * WL + x0] +
                      wx1 * gw[(z1 * HL + y0) * WL + x1g]) +
               wy1 * (wx0 * gw[(z1 * HL + y1g) * WL + x0] +
                      wx1 * gw[(z1 * HL + y1g) * WL + x1g]));
    invDen[tid] = 1.0f / fmaxf(den, 1e-8f);
  }

  // ---- stage A asynchronously: 108 cells x 128 ch, 512B/cell/wave ---------
  // (issued early so the DMA overlaps the W build below; ASYNCcnt tracked)
  for (int cell = wave; cell < NCELL; cell += 8) {
    int z = cell / 9, r = cell % 9;
    int yg = ylo + r / 3; if (yg > HL - 1) yg = HL - 1;
    int xg = xlo + r % 3; if (xg > WL - 1) xg = WL - 1;
    const float* src = gridF +
        ((((size_t)b * NB + z) * HL + yg) * WL + xg) * C_ + lane * 4;
    float* dst = A + cell * ASTR + lane * 4;
#if defined(__HIP_DEVICE_COMPILE__) && defined(__gfx1250__)
    unsigned long long gaddr = (unsigned long long)(uintptr_t)src;
    unsigned int laddr = (unsigned int)(uintptr_t)dst;   // low 32 = LDS offset
    asm volatile("global_load_async_to_lds_b128 %0, %1, off"
                 :: "v"(laddr), "v"(gaddr) : "memory");
#else
    *(float4*)dst = *(const float4*)src;
#endif
  }

  // ---- zero W, then scatter this pixel's 8 weights ------------------------
  for (int i = tid; i < 16 * NCELL * 16; i += 256) W[i] = 0.0f;
  __syncthreads();
  {
    float* Wc = W + (tid >> 4) * NCELL * 16 + (tid & 15);
    int ly1 = ly + 1, lx1 = lx + 1;
    Wc[(z0 * 9 + ly  * 3 + lx ) * 16] += wz0 * wy0 * wx0;
    Wc[(z0 * 9 + ly  * 3 + lx1) * 16] += wz0 * wy0 * wx1;
    Wc[(z0 * 9 + ly1 * 3 + lx ) * 16] += wz0 * wy1 * wx0;
    Wc[(z0 * 9 + ly1 * 3 + lx1) * 16] += wz0 * wy1 * wx1;
    Wc[(z1 * 9 + ly  * 3 + lx ) * 16] += wz1 * wy0 * wx0;
    Wc[(z1 * 9 + ly  * 3 + lx1) * 16] += wz1 * wy0 * wx1;
    Wc[(z1 * 9 + ly1 * 3 + lx ) * 16] += wz1 * wy1 * wx0;
    Wc[(z1 * 9 + ly1 * 3 + lx1) * 16] += wz1 * wy1 * wx1;
  }
#if defined(__HIP_DEVICE_COMPILE__) && defined(__gfx1250__)
  asm volatile("s_wait_asynccnt 0x0" ::: "memory");  // A writes visible
#endif
  __syncthreads();

  // ---- WMMA: each wave owns pixel-blocks {2*wave, 2*wave+1} ---------------
  const int m = lane & 15;       // D column / A row within block
  const int hi = lane >> 4;      // half-wave selector (K offset, M offset)
  #pragma unroll 1
  for (int pi = 0; pi < 2; ++pi) {
    const int pb = wave * 2 + pi;
    const float* Wp = W + pb * NCELL * 16 + m;

    // register-cache this pixel-block's 27 B fragments (54 VGPRs)
    v2f wreg[KSTEPS];
    #pragma unroll
    for (int k4 = 0; k4 < KSTEPS; ++k4) {
      int k = k4 * 4 + (hi ? 2 : 0);
      wreg[k4].x = Wp[k * 16];
      wreg[k4].y = Wp[(k + 1) * 16];
    }

    const int pcol = pb * 16 + m;           // pixel index in tile
    const float sc = invDen[pcol];
    const int pxx = x0h + (pcol & 15);
    const int pyy = y0h + (pcol >> 4);

    #pragma unroll 1
    for (int cb = 0; cb < 8; ++cb) {
      const float* Ap = A + cb * 16 + m;

      // preload all 27 A fragments as one ds_load clause (54 VGPRs), then
      // run the WMMA chain back-to-back behind a single dscnt wait.
      v2f areg[KSTEPS];
      #pragma unroll
      for (int k4 = 0; k4 < KSTEPS; ++k4) {
        int k = k4 * 4 + (hi ? 2 : 0);
        areg[k4].x = Ap[(size_t)k * ASTR];
        areg[k4].y = Ap[(size_t)(k + 1) * ASTR];
      }

      v8f acc = {0.f, 0.f, 0.f, 0.f, 0.f, 0.f, 0.f, 0.f};
#if __has_builtin(__builtin_amdgcn_wmma_f32_16x16x4_f32)
      #pragma unroll
      for (int k4 = 0; k4 < KSTEPS; ++k4) {
        acc = __builtin_amdgcn_wmma_f32_16x16x4_f32(
            /*neg_a=*/false, areg[k4], /*neg_b=*/false, wreg[k4],
            /*c_mod=*/(short)0, acc, /*reuse_a=*/false, /*reuse_b=*/false);
      }
#else
      for (int k4 = 0; k4 < KSTEPS; ++k4) {
        #pragma unroll
        for (int r = 0; r < 8; ++r)
          acc[r] += areg[k4].x * wreg[k4].x + areg[k4].y * wreg[k4].y;
      }
#endif
      // ---- store D tile, dividing by the trilerped weight sum -------------
      float* op = out +
          (((size_t)b * C_ + cb * 16 + hi * 8) * HH + pyy) * WH + pxx;
      #pragma unroll
      for (int r = 0; r < 8; ++r)
        op[(size_t)r * HH * WH] = acc[r] * sc;
    }
  }
}

// ---------------------------------------------------------------------------
extern "C" void kernel_launch(void* const* d_in, const int* in_sizes, int n_in,
                              void* d_out, int out_size, void* d_ws, size_t ws_size,
                              hipStream_t stream) {
  const float* feat   = (const float*)d_in[0];
  const float* guide  = (const float*)d_in[1];
  const float* sx_raw = (const float*)d_in[2];
  const float* sy_raw = (const float*)d_in[3];
  // d_in[4] (th_raw) is dead code in the reference.
  const float* sr_raw = (const float*)d_in[5];
  float* out = (float*)d_out;

  // workspace layout (~12.7 MB, L2-resident)
  float* luma_lr = (float*)d_ws;                       // B*32*32
  float* gridW   = luma_lr + B_ * HL * WL;             // B*12*32*32
  float* gridF   = gridW + B_ * NB * HL * WL;          // B*12*32*32*128

  luma_lr_kernel<<<(B_ * HL * WL + 255) / 256, 256, 0, stream>>>(guide, luma_lr);
  grid_kernel<<<B_ * NB * HL, C_, 0, stream>>>(feat, luma_lr, sx_raw, sy_raw,
                                               sr_raw, gridF, gridW);

  size_t smem = (size_t)(NCELL * ASTR + 16 * NCELL * 16 + 256) * sizeof(float);
  hipFuncSetAttribute((const void*)slice_kernel,
                      hipFuncAttributeMaxDynamicSharedMemorySize, (int)smem);
  slice_kernel<<<dim3(WH / 16, HH / 16, B_), 256, smem, stream>>>(
      guide, gridF, gridW, out);
}